// QuantumNeuralNetwork_50964081934623
// MI455X (gfx1250) — compile-verified
//
#include <hip/hip_runtime.h>
#include <cstdint>

#define NQ        1024      // num qubits == batch
#define IN_STRIDE 16384     // INPUT_SIZE (row stride of x)
#define HIDDEN    64
#define OUT_N     1000

typedef __attribute__((ext_vector_type(2))) float v2f;
typedef __attribute__((ext_vector_type(8))) float v8f;

// ---------------------------------------------------------------------------
// Hardware tanh (CDNA5 TRANS op v_tanh_f32)
// ---------------------------------------------------------------------------
__device__ __forceinline__ float fast_tanh(float v) {
#if __has_builtin(__builtin_amdgcn_tanhf)
    return __builtin_amdgcn_tanhf(v);
#elif __has_builtin(__builtin_amdgcn_tanh_f32)
    return __builtin_amdgcn_tanh_f32(v);
#else
    float r;
    // v_nop covers the TRANS-result hazard since the compiler can't see it
    asm volatile("v_tanh_f32 %0, %1\n\tv_nop" : "=v"(r) : "v"(v));
    return r;
#endif
}

// ---------------------------------------------------------------------------
// Threefry-2x32 (JAX PRNG), key = jax.random.key(42) -> {0, 42}
// bits layout for n elements: pair (i, i+n/2); out0 -> idx<n/2, out1 -> idx>=n/2
// ---------------------------------------------------------------------------
__device__ __forceinline__ uint32_t rotl32(uint32_t x, int r) {
    return (x << r) | (x >> (32 - r));
}

__device__ uint32_t threefry_bits(uint32_t idx) {
    const uint32_t half = (NQ * NQ) / 2u;
    bool hi = idx >= half;
    uint32_t x0 = hi ? (idx - half) : idx;
    uint32_t x1 = hi ? idx : (idx + half);
    const uint32_t k0 = 0u, k1 = 42u;
    const uint32_t k2 = 0x1BD11BDAu ^ k0 ^ k1;
    x0 += k0; x1 += k1;
#define TFR(r) { x0 += x1; x1 = rotl32(x1, r); x1 ^= x0; }
    TFR(13) TFR(15) TFR(26) TFR(6)   x0 += k1; x1 += k2 + 1u;
    TFR(17) TFR(29) TFR(16) TFR(24)  x0 += k2; x1 += k0 + 2u;
    TFR(13) TFR(15) TFR(26) TFR(6)   x0 += k0; x1 += k1 + 3u;
    TFR(17) TFR(29) TFR(16) TFR(24)  x0 += k1; x1 += k2 + 4u;
    TFR(13) TFR(15) TFR(26) TFR(6)   x0 += k2; x1 += k0 + 5u;
#undef TFR
    return hi ? x1 : x0;
}

__device__ __forceinline__ float gumbel_noise(uint32_t idx) {
    uint32_t bits = threefry_bits(idx);
    float f = __uint_as_float((bits >> 9) | 0x3F800000u) - 1.0f;  // [0,1)
    const float tiny = 1.1754944e-38f;
    float u = f * (1.0f - tiny) + tiny;                            // [tiny,1)
    return -logf(-logf(u));
}

// ---------------------------------------------------------------------------
// Kernel 1: per-row quantum scan (1024 sequential steps) + gumbel-argmax.
// 1 block per batch row; 256 threads x 4 independent chains (TRANS latency
// hiding). Coefficients folded to 2 per step and broadcast from LDS.
// ---------------------------------------------------------------------------
__global__ __launch_bounds__(256) void qnn_scan_argmax(
    const float* __restrict__ x, const float* __restrict__ rot,
    float* __restrict__ qt) {
    __shared__ float sA[NQ];   // cos(rx)*cos(rz)
    __shared__ float sB[NQ];   // sin(ry)*cos(rz)
    __shared__ float rval[256];
    __shared__ int   ridx[256];

    const int b = blockIdx.x;
    const int t = threadIdx.x;

    for (int i = t; i < NQ; i += 256) {
        float rx = rot[i * 3 + 0];
        float ry = rot[i * 3 + 1];
        float rz = rot[i * 3 + 2];
        float cz = cosf(rz);
        sA[i] = cosf(rx) * cz;
        sB[i] = sinf(ry) * cz;
    }
    __syncthreads();

    // init: s = tanh(x[b, q]) for 4 qubits per thread
    const float* xrow = x + (size_t)b * IN_STRIDE;
    float s0 = fast_tanh(xrow[t + 0 * 256]);
    float s1 = fast_tanh(xrow[t + 1 * 256]);
    float s2 = fast_tanh(xrow[t + 2 * 256]);
    float s3 = fast_tanh(xrow[t + 3 * 256]);

    // 1024 sequential quantum layers: s = fma(s, A, B*tanh(s))
#pragma unroll 2
    for (int i = 0; i < NQ; ++i) {
        float A = sA[i];
        float B = sB[i];
        float t0 = fast_tanh(s0);
        float t1 = fast_tanh(s1);
        float t2 = fast_tanh(s2);
        float t3 = fast_tanh(s3);
        s0 = fmaf(s0, A, B * t0);
        s1 = fmaf(s1, A, B * t1);
        s2 = fmaf(s2, A, B * t2);
        s3 = fmaf(s3, A, B * t3);
    }

    // categorical sample == argmax(state + gumbel) (log_softmax shift cancels)
    float bv = -3.402823466e+38f;
    int   bi = 0;
    float sv[4] = {s0, s1, s2, s3};
#pragma unroll
    for (int j = 0; j < 4; ++j) {
        int q = t + j * 256;
        float v = sv[j] + gumbel_noise((uint32_t)(b * NQ + q));
        if (v > bv) { bv = v; bi = q; }   // increasing q => first-occurrence ties
    }
    rval[t] = bv; ridx[t] = bi;
    __syncthreads();
    for (int off = 128; off > 0; off >>= 1) {
        if (t < off) {
            float vo = rval[t + off]; int io = ridx[t + off];
            if (vo > rval[t] || (vo == rval[t] && io < ridx[t])) {
                rval[t] = vo; ridx[t] = io;
            }
        }
        __syncthreads();
    }
    if (t == 0) qt[b] = (float)ridx[0];
}

// ---------------------------------------------------------------------------
// Kernel 2/3: GEMV via v_wmma_f32_16x16x4_f32 (exact f32 matrix pipe).
// One wave per 16-row tile of W [M,K] row-major. Vector sits in column N=0
// of B; result column read from C lanes 0 (M=0..7) and 16 (M=8..15).
// ---------------------------------------------------------------------------
__global__ __launch_bounds__(32) void qnn_gemv_wmma(
    const float* __restrict__ W, const float* __restrict__ bias,
    const float* __restrict__ vin, float* __restrict__ vout,
    int M, int K, int do_relu) {
    const int lane = threadIdx.x;
    const int tile = blockIdx.x;

#if __has_builtin(__builtin_amdgcn_wmma_f32_16x16x4_f32)
    const int m  = tile * 16 + (lane & 15);   // A row for this lane
    const int kb = (lane >> 4) * 2;           // K sub-offset (0 or 2)
    const int n  = lane & 15;                 // B/C column
    const bool arow = (m < M);
    v8f c = {};
    for (int k = 0; k < K; k += 4) {
        v2f a, bb;
        a.x  = arow ? W[(size_t)m * K + k + kb]     : 0.0f;
        a.y  = arow ? W[(size_t)m * K + k + kb + 1] : 0.0f;
        bb.x = (n == 0) ? vin[k + kb]     : 0.0f;
        bb.y = (n == 0) ? vin[k + kb + 1] : 0.0f;
        c = __builtin_amdgcn_wmma_f32_16x16x4_f32(
                false, a, false, bb, (short)0, c, false, false);
    }
    if (n == 0) {
        int mbase = tile * 16 + ((lane >> 4) ? 8 : 0);
#pragma unroll
        for (int j = 0; j < 8; ++j) {
            int row = mbase + j;
            if (row < M) {
                float v = c[j] + bias[row];
                vout[row] = do_relu ? fmaxf(v, 0.0f) : v;
            }
        }
    }
#else
    // Fallback (keeps the build green if the f32 WMMA builtin is absent;
    // histogram wmma=0 would flag it for the next iteration).
    int row = tile * 16 + lane;
    if (lane < 16 && row < M) {
        float acc = bias[row];
        for (int k = 0; k < K; ++k) acc = fmaf(W[(size_t)row * K + k], vin[k], acc);
        vout[row] = do_relu ? fmaxf(acc, 0.0f) : acc;
    }
#endif
}

// ---------------------------------------------------------------------------
extern "C" void kernel_launch(void* const* d_in, const int* in_sizes, int n_in,
                              void* d_out, int out_size, void* d_ws, size_t ws_size,
                              hipStream_t stream) {
    const float* x   = (const float*)d_in[0];
    const float* rot = (const float*)d_in[1];
    const float* W1  = (const float*)d_in[2];
    const float* b1  = (const float*)d_in[3];
    const float* W2  = (const float*)d_in[4];
    const float* b2  = (const float*)d_in[5];
    float* out = (float*)d_out;

    float* qt = (float*)d_ws;      // [1024]
    float* h  = qt + NQ;           // [64]

    qnn_scan_argmax<<<NQ, 256, 0, stream>>>(x, rot, qt);
    qnn_gemv_wmma<<<(HIDDEN + 15) / 16, 32, 0, stream>>>(W1, b1, qt, h, HIDDEN, NQ, 1);
    qnn_gemv_wmma<<<(OUT_N + 15) / 16, 32, 0, stream>>>(W2, b2, h, out, OUT_N, HIDDEN, 0);
}